// mLSTMBlock_58067957842089
// MI455X (gfx1250) — compile-verified
//
#include <hip/hip_runtime.h>
#include <math.h>

#define BATCH 64
#define INDIM 512
#define HDIM  1024

typedef float v2f __attribute__((ext_vector_type(2)));
typedef float v8f __attribute__((ext_vector_type(8)));
typedef float f4  __attribute__((ext_vector_type(4)));

__device__ __forceinline__ float sigmoidf_(float x) {
    return 1.0f / (1.0f + __expf(-x));
}

// ---------------------------------------------------------------------------
// Kernel 1: six gate GEMMs  out[b,h] = sum_k x[b,k] * W[h,k]   (M=64,K=512,N=1024)
// One wave computes one 16x16 output tile via V_WMMA_F32_16X16X4_F32.
// blockIdx.y selects the weight matrix (scalar-uniform -> clean s_cbranch
// epilogue, no exec-mask divergence). 256 tiles/weight = 32 blocks x 8 waves.
// ---------------------------------------------------------------------------
__global__ __launch_bounds__(256) void gates_gemm_wmma(
    const float* __restrict__ x,
    const float* __restrict__ Wq, const float* __restrict__ Wk,
    const float* __restrict__ Wv, const float* __restrict__ Wi,
    const float* __restrict__ Wf, const float* __restrict__ Wo,
    const float* __restrict__ bi, const float* __restrict__ bf,
    const float* __restrict__ bo,
    float* __restrict__ qs, float* __restrict__ kt, float* __restrict__ vt,
    float* __restrict__ it, float* __restrict__ ft, float* __restrict__ ot)
{
    const int w    = blockIdx.y;                               // 0..5, uniform
    const int tile = blockIdx.x * 8 + (threadIdx.x >> 5);      // 0..255
    const int m0   = (tile >> 6) << 4;                         // M tile base
    const int n0   = (tile & 63) << 4;                         // N tile base

    const int lane = threadIdx.x & 31;
    const int half = lane >> 4;        // 0: K pair {0,1}   1: K pair {2,3}
    const int lm   = lane & 15;

    const float* W;
    switch (w) {
        case 0:  W = Wq; break;
        case 1:  W = Wk; break;
        case 2:  W = Wv; break;
        case 3:  W = Wi; break;
        case 4:  W = Wf; break;
        default: W = Wo; break;
    }

    const float* arow = x + (size_t)(m0 + lm) * INDIM + 2 * half;
    const float* brow = W + (size_t)(n0 + lm) * INDIM + 2 * half;

    v8f c = {};
    #pragma unroll 4
    for (int k = 0; k < INDIM; k += 4) {
        v2f a = *(const v2f*)(arow + k);   // A[16x4] frag: lane-half picks K pair
        v2f b = *(const v2f*)(brow + k);   // B[4x16] frag: W^T, mirrored layout
        c = __builtin_amdgcn_wmma_f32_16x16x4_f32(
                /*neg_a=*/false, a, /*neg_b=*/false, b,
                /*c_mod=*/(short)0, c, /*reuse_a=*/false, /*reuse_b=*/false);
    }

    // D layout: VGPR r -> M = m0 + r + 8*half, N = n0 + (lane&15)
    const int n = n0 + lm;
    const size_t base = (size_t)(m0 + (half << 3)) * HDIM + n;

    switch (w) {
        case 0: {   // q * 1/sqrt(H)
            #pragma unroll
            for (int r = 0; r < 8; ++r) qs[base + (size_t)r * HDIM] = c[r] * 0.03125f;
            break;
        }
        case 1: {
            #pragma unroll
            for (int r = 0; r < 8; ++r) kt[base + (size_t)r * HDIM] = c[r];
            break;
        }
        case 2: {
            #pragma unroll
            for (int r = 0; r < 8; ++r) vt[base + (size_t)r * HDIM] = c[r];
            break;
        }
        case 3: {   // input gate: exp(. + b_i)
            const float bv = bi[n];
            #pragma unroll
            for (int r = 0; r < 8; ++r) it[base + (size_t)r * HDIM] = __expf(c[r] + bv);
            break;
        }
        case 4: {   // forget gate: sigmoid(. + b_f)
            const float bv = bf[n];
            #pragma unroll
            for (int r = 0; r < 8; ++r) ft[base + (size_t)r * HDIM] = sigmoidf_(c[r] + bv);
            break;
        }
        default: {  // output gate: sigmoid(. + b_o)
            const float bv = bo[n];
            #pragma unroll
            for (int r = 0; r < 8; ++r) ot[base + (size_t)r * HDIM] = sigmoidf_(c[r] + bv);
            break;
        }
    }
}

// ---------------------------------------------------------------------------
// Kernel 2: the HBM-bound pass (512 MB of streaming traffic ~ 22 us roofline).
//   C_t[b,v,:] = f[b,v]*C_prev[b,v,:] + (i[b,v]*vv[b,v]) * k[b,:]
//   retrieved[b,v] = dot(C_t[b,v,:], q[b,:])     (fused, free)
// One wave per row; float4 non-temporal streaming; k,q staged in LDS.
// grid = (H/8, B), block = 256 (8 waves, 8 rows).
// ---------------------------------------------------------------------------
__global__ __launch_bounds__(256) void update_C_stream(
    const float* __restrict__ C_prev, const float* __restrict__ kt,
    const float* __restrict__ qs,     const float* __restrict__ it,
    const float* __restrict__ ft,     const float* __restrict__ vt,
    float* __restrict__ C_out, float* __restrict__ retr)
{
    __shared__ float sk[HDIM];
    __shared__ float sq[HDIM];

    const int b = blockIdx.y;
    for (int i = threadIdx.x; i < HDIM; i += 256) {
        sk[i] = kt[(size_t)b * HDIM + i];
        sq[i] = qs[(size_t)b * HDIM + i];
    }
    __syncthreads();

    const int wv   = threadIdx.x >> 5;
    const int lane = threadIdx.x & 31;
    const int v    = (blockIdx.x << 3) + wv;

    const float f  = ft[(size_t)b * HDIM + v];
    const float iv = it[(size_t)b * HDIM + v] * vt[(size_t)b * HDIM + v];

    const f4* Cp = (const f4*)(C_prev + ((size_t)b * HDIM + v) * HDIM);
    f4*       Co = (f4*)      (C_out  + ((size_t)b * HDIM + v) * HDIM);
    const f4* k4 = (const f4*)sk;
    const f4* q4 = (const f4*)sq;

    float acc = 0.0f;
    #pragma unroll
    for (int step = 0; step < 8; ++step) {
        const int idx = step * 32 + lane;                 // 256 float4 = 1024 floats
        f4 cold = __builtin_nontemporal_load(Cp + idx);   // stream read (NT)
        f4 kk   = k4[idx];
        f4 cnew = f * cold + iv * kk;
        __builtin_nontemporal_store(cnew, Co + idx);      // stream write (NT)
        f4 p = cnew * q4[idx];
        acc += p.x + p.y + p.z + p.w;
    }

    // wave32 reduction of the q-dot
    #pragma unroll
    for (int off = 16; off > 0; off >>= 1)
        acc += __shfl_xor(acc, off, 32);
    if (lane == 0) retr[(size_t)b * HDIM + v] = acc;
}

// ---------------------------------------------------------------------------
// Kernel 3: n_t, normalizer reduction, h_t.  One block per batch element.
// ---------------------------------------------------------------------------
__global__ __launch_bounds__(256) void finalize_h(
    const float* __restrict__ qs, const float* __restrict__ kt,
    const float* __restrict__ it, const float* __restrict__ ft,
    const float* __restrict__ ot, const float* __restrict__ retr,
    const float* __restrict__ n_prev,
    float* __restrict__ h_out, float* __restrict__ n_out)
{
    const int b   = blockIdx.x;
    const int tid = threadIdx.x;
    __shared__ float red[256];
    __shared__ float inv_norm;

    float s = 0.0f;
    for (int h = tid; h < HDIM; h += 256) {
        const size_t o = (size_t)b * HDIM + h;
        float n = ft[o] * n_prev[o] + it[o] * kt[o];
        n_out[o] = n;
        s += n * qs[o];
    }
    red[tid] = s;
    __syncthreads();
    #pragma unroll
    for (int st = 128; st > 0; st >>= 1) {
        if (tid < st) red[tid] += red[tid + st];
        __syncthreads();
    }
    if (tid == 0) inv_norm = 1.0f / fmaxf(fabsf(red[0]), 1.0f);
    __syncthreads();

    const float inv = inv_norm;
    for (int h = tid; h < HDIM; h += 256) {
        const size_t o = (size_t)b * HDIM + h;
        h_out[o] = ot[o] * retr[o] * inv;
    }
}

// ---------------------------------------------------------------------------
// launch
// ---------------------------------------------------------------------------
extern "C" void kernel_launch(void* const* d_in, const int* in_sizes, int n_in,
                              void* d_out, int out_size, void* d_ws, size_t ws_size,
                              hipStream_t stream) {
    const float* x      = (const float*)d_in[0];
    // d_in[1] = h_prev (unused by reference)
    const float* C_prev = (const float*)d_in[2];
    const float* n_prev = (const float*)d_in[3];
    const float* Wq     = (const float*)d_in[4];
    const float* Wk     = (const float*)d_in[5];
    const float* Wv     = (const float*)d_in[6];
    const float* Wi     = (const float*)d_in[7];
    const float* Wf     = (const float*)d_in[8];
    const float* Wo     = (const float*)d_in[9];
    const float* bi     = (const float*)d_in[10];
    const float* bf     = (const float*)d_in[11];
    const float* bo     = (const float*)d_in[12];

    // output tuple layout: h_t [B,H] | C_t [B,H,H] | n_t [B,H]
    float* h_out = (float*)d_out;
    float* C_out = h_out + (size_t)BATCH * HDIM;
    float* n_out = C_out + (size_t)BATCH * HDIM * HDIM;

    // workspace: 7 * B*H floats (~1.75 MB)
    const size_t BH = (size_t)BATCH * HDIM;
    float* qs   = (float*)d_ws;
    float* kt   = qs + BH;
    float* vt   = kt + BH;
    float* it   = vt + BH;
    float* ft   = it + BH;
    float* ot   = ft + BH;
    float* retr = ot + BH;

    // 1) gates via f32 WMMA: grid (32 tiles-blocks, 6 weights), 8 waves/block
    gates_gemm_wmma<<<dim3(32, 6), dim3(256), 0, stream>>>(
        x, Wq, Wk, Wv, Wi, Wf, Wo, bi, bf, bo, qs, kt, vt, it, ft, ot);

    // 2) streaming C update + fused retrieval dot
    update_C_stream<<<dim3(HDIM / 8, BATCH), dim3(256), 0, stream>>>(
        C_prev, kt, qs, it, ft, vt, C_out, retr);

    // 3) n_t, normalizer, h_t
    finalize_h<<<dim3(BATCH), dim3(256), 0, stream>>>(
        qs, kt, it, ft, ot, retr, n_prev, h_out, n_out);
}